// MultiHeadCausalRAG_78520592105869
// MI455X (gfx1250) — compile-verified
//
#include <hip/hip_runtime.h>
#include <math.h>

// ---------------------------------------------------------------------------
// MultiHeadCausalRAG for MI455X (gfx1250, wave32, WMMA).
// Dominant op: sim = patient_emb @ l2norm(corpus).T  (128 x 262144, K=256, f32)
//  -> bandwidth-bound (268MB corpus stream @ 23.3TB/s ~ 11.5us floor).
//     Corpus is streamed once with DOUBLE-BUFFERED LDS tiles filled
//     alternately by the Tensor Data Mover (tensor_load_to_lds, TENSORcnt)
//     and per-lane async DMA (global_load_async_to_lds_b128, ASYNCcnt),
//     overlapped with doc-norm + f32 WMMA 16x16x4 + per-chunk top-8.
// All dense layers use a generic f32 WMMA GEMM (D = relu?(A @ W^T + b)).
// ---------------------------------------------------------------------------

typedef __attribute__((ext_vector_type(2))) float v2f;
typedef __attribute__((ext_vector_type(8))) float v8f;
typedef __attribute__((ext_vector_type(4))) unsigned int v4u;
typedef __attribute__((ext_vector_type(4))) int v4i;
typedef __attribute__((ext_vector_type(8))) int v8i;

#if defined(__has_builtin)
#if __has_builtin(__builtin_amdgcn_tensor_load_to_lds)
#define HAVE_TDM 1
#endif
#endif

#define B_SZ   128
#define EMB    256
#define NDOCS  262144
#define TOPK   8
#define DOCS_PER_WG 512
#define NTILES (DOCS_PER_WG / 16)      // 32
#define NCHUNK (NDOCS / DOCS_PER_WG)   // 512
#define XDIM   2112                    // CONF + EMB*TOPK = 64 + 2048

__device__ __forceinline__ void topk_insert(float* s, int* id, float val, int idx) {
  if (val <= s[TOPK - 1]) return;
  int j = TOPK - 1;
  while (j > 0 && s[j - 1] < val) { s[j] = s[j - 1]; id[j] = id[j - 1]; --j; }
  s[j] = val; id[j] = idx;
}

// ---------------- patient encoder + l2norm:  pe[128,256] ----------------
__global__ __launch_bounds__(256) void rag_patient_enc(
    const float* __restrict__ patient, const float* __restrict__ W,
    const float* __restrict__ b, float* __restrict__ pe) {
  __shared__ float prow[68];
  __shared__ float red[256];
  const int brow = blockIdx.x, tid = threadIdx.x;
  if (tid < 68) prow[tid] = patient[brow * 68 + tid];
  __syncthreads();
  float acc = b[tid];
  const float* wr = W + (size_t)tid * 68;
  #pragma unroll 4
  for (int i = 0; i < 68; ++i) acc += wr[i] * prow[i];
  red[tid] = acc * acc;
  __syncthreads();
  for (int off = 128; off; off >>= 1) {
    if (tid < off) red[tid] += red[tid + off];
    __syncthreads();
  }
  const float rinv = 1.0f / fmaxf(sqrtf(red[0]), 1e-12f);
  pe[(size_t)brow * EMB + tid] = acc * rinv;
}

// --- fused similarity: TDM/async LDS double buffering + WMMA + top-8 -------
__global__ __launch_bounds__(256) void rag_sim_topk(
    const float* __restrict__ pe, const float* __restrict__ corpus,
    float* __restrict__ pscore, int* __restrict__ pidx) {
  __shared__ float tileB[2][16 * EMB]; // ping-pong: 16 docs x 256 floats each
  __shared__ float rnorm[16];
  __shared__ float simbuf[B_SZ * 16];  // 128 rows x 16 docs
  const int lane  = threadIdx.x;           // 0..31
  const int wave  = threadIdx.y;           // 0..7  -> M tile
  const int tid   = wave * 32 + lane;
  const int col   = lane & 15;
  const int khalf = lane >> 4;             // 0/1
  const int docBase0 = blockIdx.x * DOCS_PER_WG;

  float ts[TOPK]; int ti[TOPK];
  #pragma unroll
  for (int i = 0; i < TOPK; ++i) { ts[i] = -3.4e38f; ti[i] = 0; }

  const float* arow = pe + (size_t)(wave * 16 + col) * EMB;

  // per-lane async DMA of one 16x256 f32 tile into LDS buffer `buf`
  auto issue_async = [&](int buf, int docBase) {
    const unsigned int ldsBase =
        (unsigned int)(unsigned long long)(&tileB[buf][0]); // LDS byte offset
    const unsigned long long gBase =
        (unsigned long long)(const void*)(corpus + (size_t)docBase * EMB);
    #pragma unroll
    for (int i = 0; i < 4; ++i) {
      const int f4 = tid + i * 256;                // float4 index in tile
      unsigned int ldsoff = ldsBase + (unsigned int)f4 * 16u;
      unsigned long long ga = gBase + (unsigned long long)f4 * 16ull;
      asm volatile("global_load_async_to_lds_b128 %0, %1, off"
                   :: "v"(ldsoff), "v"(ga) : "memory");
    }
  };

#ifdef HAVE_TDM
  // one TDM descriptor moves the whole 16x256 f32 tile (2-D tensor)
  auto issue_tdm = [&](int buf, int docBase) {
    const unsigned int ldsBase =
        (unsigned int)(unsigned long long)(&tileB[buf][0]);
    const unsigned long long ga =
        (unsigned long long)(const void*)(corpus + (size_t)docBase * EMB);
    v4u g0;
    g0[0] = 1u;                                          // count=1 (valid D#)
    g0[1] = ldsBase;                                     // lds_addr (bytes)
    g0[2] = (unsigned int)(ga & 0xFFFFFFFFu);            // global_addr[31:0]
    g0[3] = (unsigned int)((ga >> 32) & 0x01FFFFFFu)     // global_addr[56:32]
            | 0x80000000u;                               // type=2 ("image")
    v8i g1;
    g1[0] = 0x00020000;          // wg_mask=0, data_size=2 (4 bytes)
    g1[1] = (int)(256u << 16);   // tensor_dim0 = 256 (bits 79:48, low half)
    g1[2] = (int)(16u << 16);    // tensor_dim1 = 16  (bits 111:80, low half)
    g1[3] = (int)(256u << 16);   // tile_dim0 = 256   (bits 127:112)
    g1[4] = 16;                  // tile_dim1 = 16, tile_dim2 = 0
    g1[5] = 256;                 // tensor_dim0_stride = 256 (bits 207:160)
    g1[6] = 0;                   // stride0 hi, stride1 lo
    g1[7] = 0;                   // stride1 hi
    v4i g2 = {};                 // 2-D tensor: groups 2/3 unused
    v4i g3 = {};
#if __clang_major__ >= 23
    v8i g4 = {};
    __builtin_amdgcn_tensor_load_to_lds(g0, g1, g2, g3, g4, 0);
#else
    __builtin_amdgcn_tensor_load_to_lds(g0, g1, g2, g3, 0);
#endif
  };
#endif

  // tile fill: even tiles via TDM (wave 0 issues once), odd tiles via async
  auto issue_tile = [&](int buf, int tileIdx) {
    const int docBase = docBase0 + tileIdx * 16;
#ifdef HAVE_TDM
    if ((tileIdx & 1) == 0) {
      if (wave == 0) issue_tdm(buf, docBase);
    } else {
      issue_async(buf, docBase);
    }
#else
    issue_async(buf, docBase);
#endif
  };

  issue_tile(0, 0);  // prologue

  for (int nt = 0; nt < NTILES; ++nt) {
    const int cur = nt & 1;
    const int docBase = docBase0 + nt * 16;
#ifdef HAVE_TDM
    __builtin_amdgcn_s_wait_tensorcnt(0);             // no-op if none issued
#endif
    asm volatile("s_wait_asynccnt 0x0" ::: "memory"); // my async portion landed
    __syncthreads();                                  // tile visible to all
    if (nt + 1 < NTILES) issue_tile(cur ^ 1, nt + 1); // overlap next tile

    if (tid < 16) {
      float ss = 0.f;
      const float* dr = &tileB[cur][tid * EMB];
      #pragma unroll 4
      for (int e = 0; e < EMB; ++e) ss += dr[e] * dr[e];
      rnorm[tid] = 1.0f / fmaxf(sqrtf(ss), 1e-12f);
    }
    __syncthreads();

    // 16x16 tile: D = PE[16x256] * docs[256x16]
    v8f acc = {};
    const float* brow = &tileB[cur][col * EMB];
    #pragma unroll 8
    for (int k0 = 0; k0 < EMB; k0 += 4) {
      const int k = k0 + 2 * khalf;
      v2f a = *(const v2f*)(arow + k);
      v2f b = *(const v2f*)(brow + k);
      acc = __builtin_amdgcn_wmma_f32_16x16x4_f32(false, a, false, b,
                                                  (short)0, acc, false, false);
    }
    const float rn = rnorm[col];
    #pragma unroll
    for (int j = 0; j < 8; ++j)
      simbuf[(wave * 16 + j + 8 * khalf) * 16 + col] = acc[j] * rn;
    __syncthreads();
    if (tid < B_SZ) {
      #pragma unroll
      for (int c = 0; c < 16; ++c)
        topk_insert(ts, ti, simbuf[tid * 16 + c], docBase + c);
    }
    __syncthreads();
  }
  if (tid < B_SZ) {
    const size_t base = ((size_t)blockIdx.x * B_SZ + tid) * TOPK;
    #pragma unroll
    for (int i = 0; i < TOPK; ++i) { pscore[base + i] = ts[i]; pidx[base + i] = ti[i]; }
  }
}

// ---------------- final top-8 reduction (one block per batch row) ----------
__global__ __launch_bounds__(256) void rag_topk_reduce(
    const float* __restrict__ pscore, const int* __restrict__ pidx,
    float* __restrict__ outScores, int* __restrict__ topidx) {
  __shared__ float ls[256 * TOPK];
  __shared__ int   li[256 * TOPK];
  const int row = blockIdx.x, tid = threadIdx.x;
  float s[TOPK]; int id[TOPK];
  #pragma unroll
  for (int i = 0; i < TOPK; ++i) { s[i] = -3.4e38f; id[i] = 0; }
  for (int wg = tid; wg < NCHUNK; wg += 256) {
    const size_t base = ((size_t)wg * B_SZ + row) * TOPK;
    #pragma unroll
    for (int i = 0; i < TOPK; ++i) topk_insert(s, id, pscore[base + i], pidx[base + i]);
  }
  #pragma unroll
  for (int i = 0; i < TOPK; ++i) { ls[tid * TOPK + i] = s[i]; li[tid * TOPK + i] = id[i]; }
  __syncthreads();
  if (tid == 0) {
    float fs[TOPK]; int fi[TOPK];
    #pragma unroll
    for (int i = 0; i < TOPK; ++i) { fs[i] = -3.4e38f; fi[i] = 0; }
    for (int t = 0; t < 256; ++t)
      for (int i = 0; i < TOPK; ++i)
        topk_insert(fs, fi, ls[t * TOPK + i], li[t * TOPK + i]);
    #pragma unroll
    for (int i = 0; i < TOPK; ++i) {
      outScores[row * TOPK + i] = fs[i];
      topidx[row * TOPK + i] = fi[i];
    }
  }
}

// ---------------- gather retrieved + concat with confounders ---------------
__global__ __launch_bounds__(256) void rag_gather(
    const float* __restrict__ conf, const float* __restrict__ corpus,
    const int* __restrict__ topidx, float* __restrict__ x) {
  const int b = blockIdx.x, tid = threadIdx.x;
  if (tid < 64) x[(size_t)b * XDIM + tid] = conf[b * 64 + tid];
  for (int kk = 0; kk < TOPK; ++kk) {
    const int idx = topidx[b * TOPK + kk];
    x[(size_t)b * XDIM + 64 + kk * EMB + tid] = corpus[(size_t)idx * EMB + tid];
  }
}

// ------------- generic f32 WMMA GEMM:  C = act(A[M,K] @ W[N,K]^T + b) ------
// grid.x = N/16, block = (32,8): wave w handles M-tile w (M fixed 128).
__global__ __launch_bounds__(256) void rag_gemm_wmma(
    const float* __restrict__ A, int lda, const float* __restrict__ W, int ldw,
    const float* __restrict__ bias, float* __restrict__ C, int ldc,
    int K, int relu) {
  const int lane  = threadIdx.x;
  const int wave  = threadIdx.y;
  const int col   = lane & 15;
  const int khalf = lane >> 4;
  const int mbase = wave * 16;
  const int nbase = blockIdx.x * 16;
  const float* arow = A + (size_t)(mbase + col) * lda;
  const float* wrow = W + (size_t)(nbase + col) * ldw;
  v8f acc = {};
  #pragma unroll 8
  for (int k0 = 0; k0 < K; k0 += 4) {
    const int k = k0 + 2 * khalf;
    v2f a = *(const v2f*)(arow + k);
    v2f b = *(const v2f*)(wrow + k);
    acc = __builtin_amdgcn_wmma_f32_16x16x4_f32(false, a, false, b,
                                                (short)0, acc, false, false);
  }
  const float bv = bias[nbase + col];
  #pragma unroll
  for (int j = 0; j < 8; ++j) {
    float v = acc[j] + bv;
    if (relu) v = fmaxf(v, 0.0f);
    C[(size_t)(mbase + j + 8 * khalf) * ldc + nbase + col] = v;
  }
}

// ---------------- per-row LayerNorm (in place) -----------------------------
__global__ __launch_bounds__(256) void rag_layernorm(
    float* __restrict__ x, const float* __restrict__ g,
    const float* __restrict__ b, int N) {
  __shared__ float rs[256];
  __shared__ float rs2[256];
  const int row = blockIdx.x, tid = threadIdx.x;
  float s = 0.f, s2 = 0.f;
  for (int i = tid; i < N; i += 256) {
    const float v = x[(size_t)row * N + i];
    s += v; s2 += v * v;
  }
  rs[tid] = s; rs2[tid] = s2;
  __syncthreads();
  for (int off = 128; off; off >>= 1) {
    if (tid < off) { rs[tid] += rs[tid + off]; rs2[tid] += rs2[tid + off]; }
    __syncthreads();
  }
  const float mu  = rs[0] / (float)N;
  const float var = rs2[0] / (float)N - mu * mu;
  const float ri  = rsqrtf(var + 1e-5f);
  for (int i = tid; i < N; i += 256) {
    const float v = x[(size_t)row * N + i];
    x[(size_t)row * N + i] = (v - mu) * ri * g[i] + b[i];
  }
}

// ---------------- attention over batch axis (4 heads, d=64) ----------------
__global__ __launch_bounds__(128) void rag_attn(
    const float* __restrict__ Q, const float* __restrict__ K,
    const float* __restrict__ V, float* __restrict__ ctx) {
  __shared__ float sc[128];
  __shared__ float red[128];
  const int h = blockIdx.x >> 7, brow = blockIdx.x & 127, tid = threadIdx.x;
  const float* qv = Q + (size_t)brow * 256 + h * 64;
  const float* kv = K + (size_t)tid * 256 + h * 64;
  float s = 0.f;
  #pragma unroll 4
  for (int d = 0; d < 64; ++d) s += qv[d] * kv[d];
  s *= 0.125f; // 1/sqrt(64)
  sc[tid] = s; red[tid] = s;
  __syncthreads();
  for (int off = 64; off; off >>= 1) {
    if (tid < off) red[tid] = fmaxf(red[tid], red[tid + off]);
    __syncthreads();
  }
  const float mx = red[0];
  __syncthreads();
  const float e = __expf(sc[tid] - mx);
  sc[tid] = e; red[tid] = e;
  __syncthreads();
  for (int off = 64; off; off >>= 1) {
    if (tid < off) red[tid] += red[tid + off];
    __syncthreads();
  }
  const float inv = 1.0f / red[0];
  __syncthreads();
  if (tid < 64) {
    float acc = 0.f;
    for (int c = 0; c < 128; ++c) acc += sc[c] * V[(size_t)c * 256 + h * 64 + tid];
    ctx[(size_t)brow * 256 + h * 64 + tid] = acc * inv;
  }
}

__global__ void rag_enhance(const float* __restrict__ rep,
                            const float* __restrict__ ctx,
                            float* __restrict__ enh) {
  const int i = blockIdx.x * 256 + threadIdx.x;
  enh[i] = rep[i] + 0.1f * ctx[i];
}

__global__ void rag_colmean(const float* __restrict__ enh, float* __restrict__ m) {
  const int c = threadIdx.x;
  float s = 0.f;
  for (int b = 0; b < 128; ++b) s += enh[(size_t)b * 256 + c];
  m[c] = s * (1.0f / 128.0f);
}

// ---------------- head outputs + factual gather ----------------------------
__global__ __launch_bounds__(128) void rag_heads_out(
    const float* __restrict__ h2all, const float* __restrict__ W3,
    const float* __restrict__ b3, const float* __restrict__ treatment,
    float* __restrict__ out) {
  const int b = threadIdx.x;
  float best = treatment[b * 4]; int bi = 0;
  for (int t = 1; t < 4; ++t) {
    const float v = treatment[b * 4 + t];
    if (v > best) { best = v; bi = t; }
  }
  for (int t = 0; t < 4; ++t) {
    const float* h2 = h2all + ((size_t)t * 128 + b) * 128;
    const float* w  = W3 + t * 128;
    float acc = b3[t];
    #pragma unroll 4
    for (int i = 0; i < 128; ++i) acc += h2[i] * w[i];
    out[128 + b * 4 + t] = acc;          // counterfactual_preds
    if (t == bi) out[b] = acc;           // factual_outcome
  }
}

// ---------------- propensity head + softmax --------------------------------
__global__ __launch_bounds__(128) void rag_prop_out(
    const float* __restrict__ pbuf, const float* __restrict__ W,
    const float* __restrict__ bb, float* __restrict__ out) {
  const int b = threadIdx.x;
  float lg[4]; float mx = -3.4e38f;
  for (int t = 0; t < 4; ++t) {
    float acc = bb[t];
    const float* w = W + t * 256;
    const float* p = pbuf + (size_t)b * 256;
    #pragma unroll 4
    for (int i = 0; i < 256; ++i) acc += w[i] * p[i];
    lg[t] = acc; mx = fmaxf(mx, acc);
  }
  float sum = 0.f;
  for (int t = 0; t < 4; ++t) { lg[t] = __expf(lg[t] - mx); sum += lg[t]; }
  const float inv = 1.0f / sum;
  for (int t = 0; t < 4; ++t) out[640 + b * 4 + t] = lg[t] * inv;
}

// ---------------- ATE head on batch-mean representation --------------------
__global__ __launch_bounds__(128) void rag_ate(
    const float* __restrict__ meanb, const float* __restrict__ W1,
    const float* __restrict__ b1, const float* __restrict__ W2,
    const float* __restrict__ b2, float* __restrict__ out) {
  __shared__ float a[128];
  const int t = threadIdx.x;
  float acc = b1[t];
  const float* w = W1 + (size_t)t * 256;
  #pragma unroll 4
  for (int i = 0; i < 256; ++i) acc += w[i] * meanb[i];
  a[t] = fmaxf(acc, 0.f);
  __syncthreads();
  if (t == 0) {
    float s = b2[0];
    for (int i = 0; i < 128; ++i) s += a[i] * W2[i];
    out[2176] = s;
  }
}

// ---------------------------------------------------------------------------
extern "C" void kernel_launch(void* const* d_in, const int* in_sizes, int n_in,
                              void* d_out, int out_size, void* d_ws, size_t ws_size,
                              hipStream_t stream) {
  (void)in_sizes; (void)n_in; (void)out_size; (void)ws_size;
  // Inputs: setup_inputs() order, params flattened as jax pytree (alpha keys).
  const float* patient   = (const float*)d_in[0];
  const float* treatment = (const float*)d_in[1];
  const float* conf      = (const float*)d_in[2];
  const float* corpus    = (const float*)d_in[3];
  const float *ate1W = (const float*)d_in[4],  *ate1b = (const float*)d_in[5];
  const float *ate2W = (const float*)d_in[6],  *ate2b = (const float*)d_in[7];
  const float *e1W   = (const float*)d_in[8],  *e1b   = (const float*)d_in[9];
  const float *e2W   = (const float*)d_in[10], *e2b   = (const float*)d_in[11];
  const float *e3W   = (const float*)d_in[12], *e3b   = (const float*)d_in[13];
  const float *e4W   = (const float*)d_in[14], *e4b   = (const float*)d_in[15];
  const float *hW1   = (const float*)d_in[16], *hW2   = (const float*)d_in[17];
  const float *hW3   = (const float*)d_in[18];
  const float *hb1   = (const float*)d_in[19], *hb2   = (const float*)d_in[20];
  const float *hb3   = (const float*)d_in[21];
  const float *kW    = (const float*)d_in[22], *kb    = (const float*)d_in[23];
  const float *ln1b  = (const float*)d_in[24], *ln1g  = (const float*)d_in[25];
  const float *ln2b  = (const float*)d_in[26], *ln2g  = (const float*)d_in[27];
  const float *ln3b  = (const float*)d_in[28], *ln3g  = (const float*)d_in[29];
  const float *peW   = (const float*)d_in[30], *peb   = (const float*)d_in[31];
  const float *p1W   = (const float*)d_in[32], *p1b   = (const float*)d_in[33];
  const float *p2W   = (const float*)d_in[34], *p2b   = (const float*)d_in[35];
  const float *p3W   = (const float*)d_in[36], *p3b   = (const float*)d_in[37];
  const float *qW    = (const float*)d_in[38], *qb    = (const float*)d_in[39];
  const float *vW    = (const float*)d_in[40], *vb    = (const float*)d_in[41];
  float* out = (float*)d_out;

  // workspace layout (float offsets)
  float* w = (float*)d_ws;
  float* pe     = w;                  // 32768
  float* xbuf   = w + 32768;          // 270336
  float* e1     = w + 303104;         // 131072
  float* e2     = w + 434176;         // 65536
  float* e3     = w + 499712;         // 65536
  float* rep    = w + 565248;         // 32768
  float* Qb     = w + 598016;         // 32768
  float* Kb     = w + 630784;         // 32768
  float* Vb     = w + 663552;         // 32768
  float* ctx    = w + 696320;         // 32768
  float* enh    = w + 729088;         // 32768
  float* meanb  = w + 761856;         // 256
  float* h1b    = w + 762112;         // 32768
  float* h2all  = w + 794880;         // 65536
  float* pa     = w + 860416;         // 65536
  float* pb     = w + 925952;         // 32768
  float* pscore = w + 958720;         // 524288
  int*   pidx   = (int*)(w + 1483008);// 524288 ints
  int*   topidx = (int*)(w + 2007296);// 1024 ints

  const dim3 wblk(32, 8);

  // retrieval
  rag_patient_enc<<<128, 256, 0, stream>>>(patient, peW, peb, pe);
  rag_sim_topk<<<NCHUNK, wblk, 0, stream>>>(pe, corpus, pscore, pidx);
  rag_topk_reduce<<<128, 256, 0, stream>>>(pscore, pidx, out + 1152, topidx);
  rag_gather<<<128, 256, 0, stream>>>(conf, corpus, topidx, xbuf);

  // shared encoder
  rag_gemm_wmma<<<64, wblk, 0, stream>>>(xbuf, XDIM, e1W, XDIM, e1b, e1, 1024, XDIM, 1);
  rag_layernorm<<<128, 256, 0, stream>>>(e1, ln1g, ln1b, 1024);
  rag_gemm_wmma<<<32, wblk, 0, stream>>>(e1, 1024, e2W, 1024, e2b, e2, 512, 1024, 1);
  rag_layernorm<<<128, 256, 0, stream>>>(e2, ln2g, ln2b, 512);
  rag_gemm_wmma<<<32, wblk, 0, stream>>>(e2, 512, e3W, 512, e3b, e3, 512, 512, 1);
  rag_layernorm<<<128, 256, 0, stream>>>(e3, ln3g, ln3b, 512);
  rag_gemm_wmma<<<16, wblk, 0, stream>>>(e3, 512, e4W, 512, e4b, rep, 256, 512, 0);

  // attention
  rag_gemm_wmma<<<16, wblk, 0, stream>>>(rep, 256, qW, 256, qb, Qb, 256, 256, 0);
  rag_gemm_wmma<<<16, wblk, 0, stream>>>(treatment, 4, kW, 4, kb, Kb, 256, 4, 0);
  rag_gemm_wmma<<<16, wblk, 0, stream>>>(rep, 256, vW, 256, vb, Vb, 256, 256, 0);
  rag_attn<<<512, 128, 0, stream>>>(Qb, Kb, Vb, ctx);
  rag_enhance<<<128, 256, 0, stream>>>(rep, ctx, enh);
  rag_colmean<<<1, 256, 0, stream>>>(enh, meanb);

  // per-treatment heads
  for (int t = 0; t < 4; ++t) {
    rag_gemm_wmma<<<16, wblk, 0, stream>>>(enh, 256, hW1 + (size_t)t * 256 * 256,
                                           256, hb1 + t * 256, h1b, 256, 256, 1);
    rag_gemm_wmma<<<8, wblk, 0, stream>>>(h1b, 256, hW2 + (size_t)t * 128 * 256,
                                          256, hb2 + t * 128,
                                          h2all + (size_t)t * 128 * 128, 128, 256, 1);
  }
  rag_heads_out<<<1, 128, 0, stream>>>(h2all, hW3, hb3, treatment, out);

  // propensity
  rag_gemm_wmma<<<32, wblk, 0, stream>>>(conf, 64, p1W, 64, p1b, pa, 512, 64, 1);
  rag_gemm_wmma<<<16, wblk, 0, stream>>>(pa, 512, p2W, 512, p2b, pb, 256, 512, 1);
  rag_prop_out<<<1, 128, 0, stream>>>(pb, p3W, p3b, out);

  // ATE
  rag_ate<<<1, 128, 0, stream>>>(meanb, ate1W, ate1b, ate2W, ate2b, out);
}